// CountModule_21818433863734
// MI455X (gfx1250) — compile-verified
//
#include <hip/hip_runtime.h>
#include <math.h>

#define NP 10      // N_PROP
#define MP 36      // proposals per batch
#define TB 64      // batches (= threads) per block

typedef unsigned int u32x4 __attribute__((ext_vector_type(4)));
typedef int          i32x8 __attribute__((ext_vector_type(8)));
typedef int          i32x4 __attribute__((ext_vector_type(4)));

#if defined(__HIP_DEVICE_COMPILE__) && __has_builtin(__builtin_amdgcn_tensor_load_to_lds)
#define HAVE_TDM 1
#else
#define HAVE_TDM 0
#endif

#if HAVE_TDM
// Issue one TDM 2-D tile load: dim0 contiguous f32 per row, tile = dim0 x tile1
// rows, tensor = dim0 x tdim1 (OOB rows read as zero), row stride = stride0
// elements. Descriptor layout per CDNA5 ISA ch.8 (D#). Verified in round-1
// disasm: lowers to `tensor_load_to_lds s[g0], s[g1], s[g2], s[g3]`.
__device__ __forceinline__ void tdm_load_2d(const void* gptr, unsigned lds_off,
                                            unsigned dim0, unsigned tdim1,
                                            unsigned tile1,
                                            unsigned long long stride0) {
  unsigned long long ga = (unsigned long long)(size_t)gptr;
  // Group 0: count=1, lds_addr, 57-bit global_addr, type=2 ("image")
  u32x4 g0 = { 1u,
               lds_off,
               (unsigned)(ga & 0xFFFFFFFFull),
               (unsigned)(((ga >> 32) & 0x01FFFFFFull) | 0x80000000u) };
  // Group 1: data_size=2 (4B), tensor_dim0/1, tile_dim0/1/2, dim0 stride
  i32x8 g1 = { (int)(2u << 16),
               (int)((dim0 & 0xFFFFu) << 16),
               (int)(((dim0 >> 16) & 0xFFFFu) | ((tdim1 & 0xFFFFu) << 16)),
               (int)(((tdim1 >> 16) & 0xFFFFu) | ((dim0 & 0xFFFFu) << 16)),
               (int)(tile1 & 0xFFFFu),
               (int)(unsigned)(stride0 & 0xFFFFFFFFull),
               (int)(unsigned)((stride0 >> 32) & 0xFFFFull),
               0 };
  // Groups 2/3: benign higher dims (dim2=dim3=dim4=1, tiles unused)
  i32x4 g2 = { 1, 1, 0, 0 };
  i32x4 g3 = { 0, 0x00010000, 0, 0 };
#if __clang_major__ >= 23
  i32x8 gz = { 0, 0, 0, 0, 0, 0, 0, 0 };
  __builtin_amdgcn_tensor_load_to_lds(g0, g1, g2, g3, gz, 0);
#else
  __builtin_amdgcn_tensor_load_to_lds(g0, g1, g2, g3, 0);
#endif
}
#endif

// PiecewiseLinear.forward with precomputed cs/nw tables (17 entries each).
__device__ __forceinline__ float pwl_eval(const float* cs, const float* nw, float x) {
  float xd  = x * 16.0f;
  float fip = floorf(xd);
  int   ip  = (int)fip;
  float fr  = xd - fip;
  int i1 = ip < 16 ? ip : 16;
  int i2 = (ip + 1) < 16 ? (ip + 1) : 16;
  return cs[i1] + fr * nw[i2];
}

__global__ __launch_bounds__(TB) void count_forward_kernel(
    const float* __restrict__ boxes,   // [B,4,36]
    const float* __restrict__ attn,    // [B,36]
    const float* __restrict__ fw,      // [8,17]
    float* __restrict__ out,           // [B,11]
    int B) {
  __shared__ float sBoxes[TB * 144];   // 36 KB : 64 batches x (4*36)
  __shared__ float sAtt[TB * MP];      //  9 KB
  __shared__ float sW[8 * 17];         // raw weights
  __shared__ float sNW[8][17];         // normalized |w|
  __shared__ float sCS[8][17];         // cumsum

  const int t  = threadIdx.x;
  const int b0 = blockIdx.x * TB;
  const int b  = b0 + t;

#if HAVE_TDM
  if (t == 0) {  // one TDM issue per block (wave-level op; waves 1+ branch over)
    tdm_load_2d(fw, (unsigned)(size_t)(void*)sW, 8 * 17, 1, 1, 8 * 17);
    tdm_load_2d(boxes + (size_t)b0 * 144, (unsigned)(size_t)(void*)sBoxes,
                144, (unsigned)(B - b0), TB, 144);
    tdm_load_2d(attn + (size_t)b0 * MP, (unsigned)(size_t)(void*)sAtt,
                MP, (unsigned)(B - b0), TB, MP);
    __builtin_amdgcn_s_wait_tensorcnt(0);
  }
#else
  for (int i = t; i < 8 * 17; i += TB) sW[i] = fw[i];
  for (int i = t; i < TB * 144; i += TB) {
    long g = (long)b0 * 144 + i;
    sBoxes[i] = (g < (long)B * 144) ? boxes[g] : 0.0f;
  }
  for (int i = t; i < TB * MP; i += TB) {
    long g = (long)b0 * MP + i;
    sAtt[i] = (g < (long)B * MP) ? attn[g] : 0.0f;
  }
#endif
  __syncthreads();

  // Build PWL tables: nw = |w|/sum|w|, cs = cumsum(nw). One function per lane.
  if (t < 8) {
    float aw[17];
    float s = 0.0f;
#pragma unroll
    for (int k = 0; k < 17; ++k) { aw[k] = fabsf(sW[t * 17 + k]); s += aw[k]; }
    float c = 0.0f;
#pragma unroll
    for (int k = 0; k < 17; ++k) {
      float nwk = aw[k] / s;
      sNW[t][k] = nwk;
      c += nwk;
      sCS[t][k] = c;
    }
  }
  __syncthreads();

  if (b >= B) return;

  // ---- top-10 of 36 (set only matters; strict > == jax tie-to-lowest-index)
  float a_[MP];
#pragma unroll
  for (int m = 0; m < MP; ++m) a_[m] = sAtt[t * MP + m];

  float att[NP];
  int   idsel[NP];
  unsigned long long used = 0ull;
#pragma unroll
  for (int k = 0; k < NP; ++k) {
    float best = -3.402823466e38f;
    int   bi   = 0;
#pragma unroll
    for (int m = 0; m < MP; ++m) {
      bool freem = ((used >> m) & 1ull) == 0ull;
      bool gt    = freem && (a_[m] > best);
      best = gt ? a_[m] : best;
      bi   = gt ? m : bi;
    }
    used |= (1ull << bi);
    idsel[k] = bi;
    att[k]   = 1.0f / (1.0f + expf(-best));  // sigmoid
  }

  // ---- gather boxes + areas
  const float* myBox = &sBoxes[t * 144];
  float y1[NP], x1[NP], y2[NP], x2[NP], ar[NP];
#pragma unroll
  for (int k = 0; k < NP; ++k) {
    int m = idsel[k];
    y1[k] = myBox[m];
    x1[k] = myBox[36 + m];
    y2[k] = myBox[72 + m];
    x2[k] = myBox[108 + m];
    ar[k] = fmaxf(y2[k] - y1[k], 0.0f) * fmaxf(x2[k] - x1[k], 0.0f);
  }

  // ---- pass 1: s_i = k2 * sum_j f2(1-|att_i-att_j|), k2 = f2(1)^10
  // (score_diff == 0 in the reference, so the prod term is f2(1.0)^10).
  // f2 pair matrix is symmetric with constant diagonal f2(1): 45 evals.
  const float* cs2 = sCS[2];
  const float* nw2 = sNW[2];
  float f21 = cs2[16];                         // f2(1.0)
  float p2 = f21 * f21, p4 = p2 * p2, p8 = p4 * p4;
  float k2 = p8 * p2;                          // f2(1.0)^10
  float s_i[NP];
#pragma unroll
  for (int i = 0; i < NP; ++i) s_i[i] = f21;   // diagonal term
#pragma unroll
  for (int i = 0; i < NP; ++i) {
#pragma unroll
    for (int j = i + 1; j < NP; ++j) {
      float v = pwl_eval(cs2, nw2, 1.0f - fabsf(att[i] - att[j]));
      s_i[i] += v;
      s_i[j] += v;
    }
  }
  float r_i[NP];
#pragma unroll
  for (int i = 0; i < NP; ++i) {
    s_i[i] *= k2;
    r_i[i] = 1.0f / s_i[i];                    // 10 precise divides total
  }

  float corr_sum = 0.0f;
#pragma unroll
  for (int i = 0; i < NP; ++i)
    corr_sum += pwl_eval(sCS[0], sNW[0], att[i] * att[i]) * r_i[i];

  // ---- pass 2 (symmetric, upper triangle, weight 2 off-diagonal)
  float score_sum = 0.0f, pd_sum = 0.0f;
#pragma unroll
  for (int i = 0; i < NP; ++i) {
#pragma unroll
    for (int j = i; j < NP; ++j) {
      float t0 = fmaxf(y1[i], y1[j]);
      float t1 = fmaxf(x1[i], x1[j]);
      float u0 = fminf(y2[i], y2[j]);
      float u1 = fminf(x2[i], x2[j]);
      float inter = fmaxf(u0 - t0, 0.0f) * fmaxf(u1 - t1, 0.0f);
      float iou   = inter / (ar[i] + ar[j] - inter + 1e-12f);
      float dm    = 1.0f - iou;
      float av    = att[i] * att[j];
      float sc = pwl_eval(sCS[0], sNW[0], av) * pwl_eval(sCS[1], sNW[1], dm) *
                 (r_i[i] * r_i[j]);
      float pd = fabsf(pwl_eval(sCS[6], sNW[6], dm) - 0.5f);
      float w = (i == j) ? 1.0f : 2.0f;        // 2*x == x+x exactly in IEEE
      score_sum += w * sc;
      pd_sum    += w * pd;
    }
  }

  float pa_sum = 0.0f;
#pragma unroll
  for (int i = 0; i < NP; ++i)
    pa_sum += fabsf(pwl_eval(sCS[5], sNW[5], att[i]) - 0.5f);

  float gate = pwl_eval(sCS[7], sNW[7], pa_sum * 0.1f + pd_sum * 0.01f);

  float modE = score_sum + corr_sum;
  float c    = sqrtf(modE + 1e-20f);
  float cc   = fminf(fmaxf(c, 0.0f), 10.0f);
  float fip  = floorf(cc);
  int   ip   = (int)fip;
  float fr   = cc - fip;
  int li = ip < 10 ? ip : 10;
  int ri = (ip + 1) < 10 ? (ip + 1) : 10;

  float* o = out + (size_t)b * (NP + 1);
#pragma unroll
  for (int k = 0; k <= NP; ++k) {
    float v = ((k == li) ? (1.0f - fr) : 0.0f) + ((k == ri) ? fr : 0.0f);
    o[k] = gate * v;
  }
}

extern "C" void kernel_launch(void* const* d_in, const int* in_sizes, int n_in,
                              void* d_out, int out_size, void* d_ws, size_t ws_size,
                              hipStream_t stream) {
  const float* boxes = (const float*)d_in[0];
  const float* attn  = (const float*)d_in[1];
  const float* fw    = (const float*)d_in[2];
  float*       out   = (float*)d_out;
  int B    = in_sizes[1] / MP;
  int grid = (B + TB - 1) / TB;
  count_forward_kernel<<<grid, TB, 0, stream>>>(boxes, attn, fw, out, B);
}